// OddLayer_1821066133815
// MI455X (gfx1250) — compile-verified
//
#include <hip/hip_runtime.h>
#include <hip/hip_bf16.h>

// LDPC OddLayer on MI455X (gfx1250, wave32).
//
// out[b,e] = x_in[b, e>>2] + sum_{j in 4-group(e), j != e} x_prev[b, j]
//
// Implemented as the exact structured masked GEMM via V_WMMA_F32_16X16X4_F32:
// per 16(batch) x 16(edge) tile:
//   4 WMMAs: A = x_prev[16x4 K-chunk], B = (J4 - I4) placed in columns 4k..4k+3
//   1 WMMA : A = x_in[16x4],           B = broadcast mask (B[k][n] = (n>>2 == k))
// Full fp32 precision; memory-bound (~150 MB total @ 23.3 TB/s).

typedef float v2f __attribute__((ext_vector_type(2)));
typedef float v8f __attribute__((ext_vector_type(8)));

#define BATCH   2048
#define VNODES  2048
#define DEG     4
#define EDGES   (VNODES * DEG)   // 8192
#define BT_ITER 4                // batch tiles per wave

__global__ __launch_bounds__(256)
void ldpc_odd_wmma_kernel(const float* __restrict__ xin,
                          const float* __restrict__ xprev,
                          float* __restrict__ out)
{
    const int lane = threadIdx.x & 31;
    const int wave = threadIdx.x >> 5;      // 0..7, one 16-edge tile each
    const int h    = lane >> 4;             // half-wave select (K pair / M+8)
    const int n    = lane & 15;             // local column (edge) for B/C/D
    const int m    = lane & 15;             // local row (batch) for A loads

    const int e0 = blockIdx.x * 128 + wave * 16;  // first edge of this tile
    const int v0 = e0 >> 2;                       // first v-node (4 per tile)

    const int ng = n >> 2;   // group index of column n within tile
    const int nl = n & 3;    // position within group

    // --- Constant B operands (loop-invariant) -----------------------------
    // B layout for 4x16 fp32 (rows striped across lanes within a VGPR):
    //   lane (h, n): VGPR0 = B[2h][n], VGPR1 = B[2h+1][n]
    // Chunk k covers global K-rows e0+4k..e0+4k+3:
    //   B_k[kk][n] = (n>>2 == k) && ((n&3) != kk)      (J4 - I4 block)
    v2f bk0, bk1, bk2, bk3;
    {
        const int r0 = 2 * h, r1 = 2 * h + 1;
        bk0.x = (ng == 0 && nl != r0) ? 1.0f : 0.0f;
        bk0.y = (ng == 0 && nl != r1) ? 1.0f : 0.0f;
        bk1.x = (ng == 1 && nl != r0) ? 1.0f : 0.0f;
        bk1.y = (ng == 1 && nl != r1) ? 1.0f : 0.0f;
        bk2.x = (ng == 2 && nl != r0) ? 1.0f : 0.0f;
        bk2.y = (ng == 2 && nl != r1) ? 1.0f : 0.0f;
        bk3.x = (ng == 3 && nl != r0) ? 1.0f : 0.0f;
        bk3.y = (ng == 3 && nl != r1) ? 1.0f : 0.0f;
    }
    // Broadcast mask for x_in: B_in[kk][n] = (n>>2 == kk)
    v2f bin;
    bin.x = (ng == 2 * h    ) ? 1.0f : 0.0f;
    bin.y = (ng == 2 * h + 1) ? 1.0f : 0.0f;

    for (int it = 0; it < BT_ITER; ++it) {
        const int b0 = (blockIdx.y * BT_ITER + it) * 16;

        // A-matrix loads. fp32 A 16x4 layout: lane m -> M=m;
        //   VGPR0 = A[m, 2h], VGPR1 = A[m, 2h+1]
        // => one b64 load per K-chunk at column 4k + 2h. Lanes m and m+16
        //    jointly cover 16 contiguous bytes of each row.
        const float* prow = xprev + (size_t)(b0 + m) * EDGES  + e0;
        const float* irow = xin   + (size_t)(b0 + m) * VNODES + v0;

        v2f ain = *(const v2f*)(irow + 2 * h);
        v2f a0  = *(const v2f*)(prow +  0 + 2 * h);
        v2f a1  = *(const v2f*)(prow +  4 + 2 * h);
        v2f a2  = *(const v2f*)(prow +  8 + 2 * h);
        v2f a3  = *(const v2f*)(prow + 12 + 2 * h);

        v8f c = {};
        c = __builtin_amdgcn_wmma_f32_16x16x4_f32(false, ain, false, bin, (short)0, c, false, false);
        c = __builtin_amdgcn_wmma_f32_16x16x4_f32(false, a0,  false, bk0, (short)0, c, false, false);
        c = __builtin_amdgcn_wmma_f32_16x16x4_f32(false, a1,  false, bk1, (short)0, c, false, false);
        c = __builtin_amdgcn_wmma_f32_16x16x4_f32(false, a2,  false, bk2, (short)0, c, false, false);
        c = __builtin_amdgcn_wmma_f32_16x16x4_f32(false, a3,  false, bk3, (short)0, c, false, false);

        // D layout: VGPR r -> row b0 + 8h + r, col e0 + n
        float* orow = out + (size_t)(b0 + 8 * h) * EDGES + e0 + n;
        #pragma unroll
        for (int r = 0; r < 8; ++r) {
            orow[(size_t)r * EDGES] = c[r];
        }
    }
}

extern "C" void kernel_launch(void* const* d_in, const int* in_sizes, int n_in,
                              void* d_out, int out_size, void* d_ws, size_t ws_size,
                              hipStream_t stream) {
    (void)in_sizes; (void)n_in; (void)d_ws; (void)ws_size; (void)out_size;
    const float* x_in   = (const float*)d_in[0];   // [BATCH, VNODES]
    const float* x_prev = (const float*)d_in[1];   // [BATCH, EDGES]
    // d_in[2] (input_mask) and d_in[3] (prev_mask) carry the fixed Tanner
    // structure, which is hardcoded into the WMMA B operands above.
    float* out = (float*)d_out;                    // [BATCH, EDGES]

    dim3 block(256, 1, 1);                                   // 8 waves
    dim3 grid(EDGES / 128, BATCH / (16 * BT_ITER), 1);       // (64, 32)
    ldpc_odd_wmma_kernel<<<grid, block, 0, stream>>>(x_in, x_prev, out);
}